// GCNNet_635655160271
// MI455X (gfx1250) — compile-verified
//
#include <hip/hip_runtime.h>
#include <math.h>
#include <stdint.h>

#define FVDIM 128            // feature width of every layer in this net
#define PSTRIDE 132          // LDS row stride in float2 (128 data + 4 pad = 1056B -> +8 banks/row)

typedef float v2f __attribute__((ext_vector_type(2)));
typedef float v8f __attribute__((ext_vector_type(8)));
typedef unsigned int u32x4 __attribute__((ext_vector_type(4)));
typedef int i32x4 __attribute__((ext_vector_type(4)));
typedef int i32x8 __attribute__((ext_vector_type(8)));

// ---------------------------------------------------------------------------
// degree accumulation: deg_out[src[e]] += 1, deg_in[dst[e]] += 1  (fp32 atomics)
// ---------------------------------------------------------------------------
__global__ void degree_kernel(const int* __restrict__ src, const int* __restrict__ dst,
                              float* __restrict__ deg_out, float* __restrict__ deg_in, int E) {
    int e = blockIdx.x * blockDim.x + threadIdx.x;
    if (e >= E) return;
    unsafeAtomicAdd(&deg_out[src[e]], 1.0f);
    unsafeAtomicAdd(&deg_in[dst[e]], 1.0f);
}

// norm = max(deg,1)^-0.5, in place over 2N entries (norm_src then norm_dst)
__global__ void norm_kernel(float* __restrict__ d, int n) {
    int i = blockIdx.x * blockDim.x + threadIdx.x;
    if (i >= n) return;
    d[i] = rsqrtf(fmaxf(d[i], 1.0f));
}

// ---------------------------------------------------------------------------
// pack W (128x128 row-major) into K-pair-interleaved form:
//   Wp[k2*128 + n] = (W[2*k2][n], W[2*k2+1][n])
// so a WMMA B-frag is a single contiguous 8B element -> one ds_load_b64.
// ---------------------------------------------------------------------------
__global__ void pack_w_kernel(const float* __restrict__ W, float2* __restrict__ Wp) {
    int i = blockIdx.x * blockDim.x + threadIdx.x;          // 64*128 pairs
    if (i >= 64 * FVDIM) return;
    int k2 = i >> 7, n = i & 127;
    Wp[i] = make_float2(W[(2 * k2) * FVDIM + n], W[(2 * k2 + 1) * FVDIM + n]);
}

// ---------------------------------------------------------------------------
// scatter: m[dst[e]][:] += h[src[e]][:] * norm_src[src[e]]
// one wave32 per edge, each lane owns a float4 (4 atomic f32 adds).
// h and m both fit in the 192MB L2, so this runs at L2 atomic throughput.
// ---------------------------------------------------------------------------
__global__ void scatter_kernel(const float* __restrict__ h, const float* __restrict__ norm_src,
                               const int* __restrict__ src, const int* __restrict__ dst,
                               float* __restrict__ m, int E) {
    int wave = blockIdx.x * (blockDim.x >> 5) + (threadIdx.x >> 5);
    if (wave >= E) return;
    int lane = threadIdx.x & 31;
    int s = src[wave];
    int d = dst[wave];
    float ns = norm_src[s];
    float4 v = reinterpret_cast<const float4*>(h + (size_t)s * FVDIM)[lane];
    float* mrow = m + (size_t)d * FVDIM + lane * 4;
    unsafeAtomicAdd(mrow + 0, v.x * ns);
    unsafeAtomicAdd(mrow + 1, v.y * ns);
    unsafeAtomicAdd(mrow + 2, v.z * ns);
    unsafeAtomicAdd(mrow + 3, v.w * ns);
}

// ---------------------------------------------------------------------------
// out[r][c] = act( nd[r] * (A @ W)[r][c] + bias[c] ),  A: nrows x 128
// WMMA f32 16x16x4, 8 waves/block, each wave -> 16x128 output tile.
// Wp (pair-interleaved W) staged into LDS by the Tensor Data Mover:
//   2D tile 128x64 of 8B elements, 32B LDS pad per 1024B row
//   (pad_interval=7 -> 256 dwords, pad_amount=7 -> 8 dwords) == PSTRIDE layout.
// ---------------------------------------------------------------------------
__launch_bounds__(256)
__global__ void gcn_gemm_wmma(const float2* __restrict__ Wp, const float* __restrict__ A,
                              const float* __restrict__ nd, const float* __restrict__ bias,
                              float* __restrict__ out, int nrows, int do_elu) {
    extern __shared__ float2 Plds[];                 // 64 rows x PSTRIDE float2
    const int tid = threadIdx.x;

#if __has_builtin(__builtin_amdgcn_tensor_load_to_lds)
    if (tid < 32) {                                  // one wave issues the TDM op
        uint64_t ga  = (uint64_t)(uintptr_t)Wp;
        uint32_t lds = (uint32_t)(uintptr_t)Plds;    // LDS byte offset (low 32 bits)
        u32x4 g0;
        g0[0] = 1u;                                   // count=1, user mode, no gather
        g0[1] = lds;                                  // lds_addr
        g0[2] = (uint32_t)ga;                         // global_addr[31:0]
        g0[3] = (uint32_t)(ga >> 32) | (2u << 30);    // global_addr[56:32] | type=2
        i32x8 g1;
        g1[0] = (3 << 16) | (1 << 20) | (7 << 22) | (7 << 25); // data_size=8B, pad_en, ivl=1024B, amt=32B
        g1[1] = (128 << 16);                          // tensor_dim0 = 128 (8B units)
        g1[2] = (64 << 16);                           // tensor_dim1 = 64
        g1[3] = (128 << 16);                          // tile_dim0 = 128
        g1[4] = 64;                                   // tile_dim1 = 64 (tile_dim2 = 0)
        g1[5] = 128;                                  // tensor_dim0_stride = 128
        g1[6] = 0;
        g1[7] = 0;
        i32x4 gz = {};                                // groups 2/3 unused (<=2D)
        i32x8 g4 = {};                                // VADDR4 slot: unused, zero/NULL
        __builtin_amdgcn_tensor_load_to_lds(g0, g1, gz, gz, g4, 0);
        __builtin_amdgcn_s_wait_tensorcnt(0);
    }
#else
    for (int i = tid; i < 64 * FVDIM; i += 256) {
        int r = i >> 7, c = i & 127;
        Plds[r * PSTRIDE + c] = Wp[i];
    }
#endif
    __syncthreads();

    const int wid  = tid >> 5;
    const int lane = tid & 31;
    const int m    = lane & 15;
    const int hi   = lane >> 4;
    const int rowbase = (blockIdx.x * 8 + wid) * 16;
    if (rowbase + 16 > nrows) return;                // wave-uniform guard (EXEC all-ones)

    // A-frag (ISA 7.12.2, 32-bit A 16x4): lane = M, K pair {2hi, 2hi+1}
    const float* Arow = A + (size_t)(rowbase + m) * FVDIM + 2 * hi;
    // B-frag: lane = N (=m), pair row (k0/2 + hi): single ds_load_b64
    const float2* Pbase = Plds + (size_t)hi * PSTRIDE + m;

    v8f acc[8] = {};
    v2f a_cur = *reinterpret_cast<const v2f*>(Arow);  // k0 = 0

#pragma unroll 4
    for (int k0 = 0; k0 < FVDIM; k0 += 4) {
        v2f a_next = (k0 + 4 < FVDIM) ? *reinterpret_cast<const v2f*>(Arow + k0 + 4) : a_cur;
        const float2* Pk = Pbase + (k0 >> 1) * PSTRIDE;
#pragma unroll
        for (int nt = 0; nt < 8; ++nt) {
            v2f b = *reinterpret_cast<const v2f*>(Pk + nt * 16); // (W[k][c], W[k+1][c])
            acc[nt] = __builtin_amdgcn_wmma_f32_16x16x4_f32(
                false, a_cur, false, b, (short)0, acc[nt], false, false);
        }
        a_cur = a_next;
    }

    // epilogue: scale by norm_dst (folded out of the GEMM), bias, optional ELU
#pragma unroll
    for (int v = 0; v < 8; ++v) {
        int r = rowbase + v + 8 * hi;
        float ndv = nd[r];
#pragma unroll
        for (int nt = 0; nt < 8; ++nt) {
            int c = nt * 16 + m;
            float val = fmaf(ndv, acc[nt][v], bias[c]);
            if (do_elu) val = (val > 0.0f) ? val : expm1f(val);
            out[(size_t)r * FVDIM + c] = val;
        }
    }
}

// ---------------------------------------------------------------------------
// n_out = n_embed @ W_out + b_out   (128 -> 2), one thread per row
// ---------------------------------------------------------------------------
__global__ void head_kernel(const float* __restrict__ emb, const float* __restrict__ Wo,
                            const float* __restrict__ bo, float* __restrict__ out, int N) {
    int r = blockIdx.x * blockDim.x + threadIdx.x;
    if (r >= N) return;
    const float4* row = reinterpret_cast<const float4*>(emb + (size_t)r * FVDIM);
    float a0 = 0.0f, a1 = 0.0f;
#pragma unroll 8
    for (int k4 = 0; k4 < 32; ++k4) {
        float4 v = row[k4];
        int k = k4 * 4;
        a0 += v.x * Wo[(k + 0) * 2 + 0] + v.y * Wo[(k + 1) * 2 + 0]
            + v.z * Wo[(k + 2) * 2 + 0] + v.w * Wo[(k + 3) * 2 + 0];
        a1 += v.x * Wo[(k + 0) * 2 + 1] + v.y * Wo[(k + 1) * 2 + 1]
            + v.z * Wo[(k + 2) * 2 + 1] + v.w * Wo[(k + 3) * 2 + 1];
    }
    out[(size_t)r * 2 + 0] = a0 + bo[0];
    out[(size_t)r * 2 + 1] = a1 + bo[1];
}

// ---------------------------------------------------------------------------
extern "C" void kernel_launch(void* const* d_in, const int* in_sizes, int n_in,
                              void* d_out, int out_size, void* d_ws, size_t ws_size,
                              hipStream_t stream) {
    const float* x  = (const float*)d_in[0];
    const int* src  = (const int*)d_in[1];
    const int* dst  = (const int*)d_in[2];
    const float* W1 = (const float*)d_in[3];
    const float* b1 = (const float*)d_in[4];
    const float* W2 = (const float*)d_in[5];
    const float* b2 = (const float*)d_in[6];
    const float* W3 = (const float*)d_in[7];
    const float* b3 = (const float*)d_in[8];
    const float* Wo = (const float*)d_in[9];
    const float* bo = (const float*)d_in[10];

    const int N = in_sizes[0] / FVDIM;
    const int E = in_sizes[1];

    float* out   = (float*)d_out;
    float* n_out = out;                       // N x 2
    float* n_emb = out + (size_t)N * 2;       // N x 128

    // workspace: m-buffer, h-buffer, norms, packed weights
    float*  bufM     = (float*)d_ws;
    float*  bufH     = bufM + (size_t)N * FVDIM;
    float*  norms    = bufH + (size_t)N * FVDIM;   // 2N: [norm_src | norm_dst]
    float*  norm_src = norms;
    float*  norm_dst = norms + N;
    float2* Wp       = (float2*)(norms + 2 * (size_t)N);  // 64*128 float2 (64KB)

    const size_t hbytes   = sizeof(float) * (size_t)N * FVDIM;
    const int n_tiles     = (N + 15) / 16;
    const int gemm_blocks = (n_tiles + 7) / 8;
    const size_t lds_bytes = 64 * PSTRIDE * sizeof(float2);   // 67584 B
    const int scat_blocks = (E + 7) / 8;          // one wave per edge, 8 waves/block
    const int pack_blocks = (64 * FVDIM + 255) / 256;

    // degrees -> norms
    (void)hipMemsetAsync(norms, 0, sizeof(float) * 2 * (size_t)N, stream);
    degree_kernel<<<(E + 255) / 256, 256, 0, stream>>>(src, dst, norm_src, norm_dst, E);
    norm_kernel<<<(2 * N + 255) / 256, 256, 0, stream>>>(norms, 2 * N);

    // layer 1: x -> bufH (ELU)
    (void)hipMemsetAsync(bufM, 0, hbytes, stream);
    scatter_kernel<<<scat_blocks, 256, 0, stream>>>(x, norm_src, src, dst, bufM, E);
    pack_w_kernel<<<pack_blocks, 256, 0, stream>>>(W1, Wp);
    gcn_gemm_wmma<<<gemm_blocks, 256, lds_bytes, stream>>>(Wp, bufM, norm_dst, b1, bufH, N, 1);

    // layer 2: bufH -> bufH (ELU)
    (void)hipMemsetAsync(bufM, 0, hbytes, stream);
    scatter_kernel<<<scat_blocks, 256, 0, stream>>>(bufH, norm_src, src, dst, bufM, E);
    pack_w_kernel<<<pack_blocks, 256, 0, stream>>>(W2, Wp);
    gcn_gemm_wmma<<<gemm_blocks, 256, lds_bytes, stream>>>(Wp, bufM, norm_dst, b2, bufH, N, 1);

    // layer 3: bufH -> n_emb (no activation), written straight into d_out
    (void)hipMemsetAsync(bufM, 0, hbytes, stream);
    scatter_kernel<<<scat_blocks, 256, 0, stream>>>(bufH, norm_src, src, dst, bufM, E);
    pack_w_kernel<<<pack_blocks, 256, 0, stream>>>(W3, Wp);
    gcn_gemm_wmma<<<gemm_blocks, 256, lds_bytes, stream>>>(Wp, bufM, norm_dst, b3, n_emb, N, 0);

    // output head
    head_kernel<<<(N + 255) / 256, 256, 0, stream>>>(n_emb, Wo, bo, n_out, N);
}